// Labelsim_46866683134130
// MI455X (gfx1250) — compile-verified
//
#include <hip/hip_runtime.h>
#include <math.h>

#define EPSN 1e-12f

typedef __attribute__((ext_vector_type(2))) float v2f;
typedef __attribute__((ext_vector_type(8))) float v8f;

#if defined(__gfx1250__) && __has_builtin(__builtin_amdgcn_wmma_f32_16x16x4_f32)
#define USE_WMMA 1
#else
#define USE_WMMA 0
#endif

// ---------------------------------------------------------------------------
// Prep: normalize centers into zero-padded c_pad[16][128] (row-major, f32),
// zero the loss accumulator, copy centerf to out[1..C*D].
// Launch: 1 block, 128 threads.
// ---------------------------------------------------------------------------
__global__ void labelsim_prep_kernel(const float* __restrict__ centerf,
                                     float* __restrict__ c_pad,
                                     float* __restrict__ out,
                                     int C, int D /* == 128 */) {
  __shared__ float red[128];
  const int t = threadIdx.x;
  for (int r = 0; r < 16; ++r) {
    float v = (r < C) ? centerf[r * D + t] : 0.0f;
    red[t] = v * v;
    __syncthreads();
    for (int s = 64; s > 0; s >>= 1) {
      if (t < s) red[t] += red[t + s];
      __syncthreads();
    }
    float nrm = fmaxf(sqrtf(red[0]), EPSN);
    __syncthreads();  // everyone done reading red[0] before next overwrite
    c_pad[r * D + t] = (r < C) ? (v / nrm) : 0.0f;
  }
  for (int i = t; i < C * D; i += blockDim.x) out[1 + i] = centerf[i];
  if (t == 0) out[0] = 0.0f;
}

// ---------------------------------------------------------------------------
// Main: per-wave 16-point tiles. sims(16x16) = X(16x128) * Cn^T(128x16) via
// 32 chained V_WMMA_F32_16X16X4_F32 (full f32). ||x||^2 rides along in VALU.
// Block = 256 threads = 8 waves. Persistent grid-stride over tiles of both
// branches (xA then xB).
// ---------------------------------------------------------------------------
__global__ void __launch_bounds__(256)
labelsim_main_kernel(const float* __restrict__ xA, const float* __restrict__ xB,
                     const int* __restrict__ labA, const int* __restrict__ labB,
                     const float* __restrict__ c_pad,
                     float* __restrict__ out, int N) {
  const int lane = threadIdx.x & 31;
  const int wave = threadIdx.x >> 5;           // 0..7
  const int half = lane >> 4;                  // 0: K 0-1, 1: K 2-3 (per 4-wide step)
  const int r    = lane & 15;                  // matrix row (A) / col (B) index
  const int wgid   = blockIdx.x * (blockDim.x >> 5) + wave;
  const int nwaves = gridDim.x * (blockDim.x >> 5);

  const int tilesPer = (N + 15) >> 4;
  const int tilesTot = 2 * tilesPer;
  float local = 0.0f;

#if USE_WMMA
  __shared__ float simbuf[8][256];             // per-wave 16x16 sims

  // B fragments: lane holds Cn^T[k][n] with n = r, k = 4*kk + half*2 + {0,1}.
  // c_pad is [n][k] row-major -> contiguous b64 per lane; resident for all tiles.
  v2f Bf[32];
#pragma unroll
  for (int kk = 0; kk < 32; ++kk)
    Bf[kk] = *(const v2f*)(c_pad + r * 128 + kk * 4 + half * 2);

  for (int t = wgid; t < tilesTot; t += nwaves) {
    const int branch = (t >= tilesPer);
    const float* x   = branch ? xB : xA;
    const int* lab   = branch ? labB : labA;
    const int tt     = branch ? (t - tilesPer) : t;
    const int base   = tt << 4;

    int rowIdx = base + r;                     // tail: clamp read, mask later
    if (rowIdx >= N) rowIdx = N - 1;
    const float* rowp = x + (size_t)rowIdx * 128 + half * 2;

    v8f acc = {};
    float ps = 0.0f;                           // partial ||x||^2 (this lane's K slots)
#pragma unroll
    for (int kk = 0; kk < 32; ++kk) {
      v2f a = *(const v2f*)(rowp + kk * 4);    // A[m=r][k], b64 load
      ps = fmaf(a.x, a.x, fmaf(a.y, a.y, ps));
      acc = __builtin_amdgcn_wmma_f32_16x16x4_f32(
          /*neg_a=*/false, a, /*neg_b=*/false, Bf[kk],
          /*c_mod=*/(short)0, acc, /*reuse_a=*/false, /*reuse_b=*/false);
    }
    // full ||x||^2 for row r: combine half 0 (K%4 in {0,1}) and half 1 ({2,3})
    float ss = ps + __shfl_xor(ps, 16, 32);

    // D layout: VGPR g holds sims[m = g + 8*half][n = r]
    __builtin_amdgcn_wave_barrier();
#pragma unroll
    for (int g = 0; g < 8; ++g)
      simbuf[wave][(g + 8 * half) * 16 + r] = acc[g];
    __builtin_amdgcn_wave_barrier();           // DS in-order per wave; fence compiler

    float contrib = 0.0f;
    if (half == 0) {                           // lane r owns point r of the tile
      const int pt = base + r;
      const int L  = (pt < N) ? lab[pt] : 0;
      if (L >= 1) {
        float sim = simbuf[wave][r * 16 + L];
        contrib = 1.0f - sim / fmaxf(sqrtf(ss), EPSN);
      }
    }
    __builtin_amdgcn_wave_barrier();
    local += contrib;
  }
#else
  // Fallback (no f32 WMMA builtin): one point per wave-cooperative pass.
  for (int t = wgid; t < tilesTot; t += nwaves) {
    const int branch = (t >= tilesPer);
    const float* x   = branch ? xB : xA;
    const int* lab   = branch ? labB : labA;
    const int tt     = branch ? (t - tilesPer) : t;
    for (int p = tt * 16; p < min(tt * 16 + 16, N); ++p) {
      const int L = lab[p];
      // lane-parallel over dims: 4 floats per lane
      const float* row = x + (size_t)p * 128 + lane * 4;
      float ss = 0.0f, dt = 0.0f;
      const float* c = c_pad + L * 128 + lane * 4;
      for (int j = 0; j < 4; ++j) {
        float v = row[j];
        ss = fmaf(v, v, ss);
        dt = fmaf(v, c[j], dt);
      }
      for (int m = 16; m >= 1; m >>= 1) {
        ss += __shfl_xor(ss, m, 32);
        dt += __shfl_xor(dt, m, 32);
      }
      if (lane == 0 && L >= 1) local += 1.0f - dt / fmaxf(sqrtf(ss), EPSN);
    }
  }
#endif

  // wave reduction then one atomic per wave
  for (int m = 16; m >= 1; m >>= 1) local += __shfl_xor(local, m, 32);
  if (lane == 0) atomicAdd(out, local);
}

// ---------------------------------------------------------------------------
extern "C" void kernel_launch(void* const* d_in, const int* in_sizes, int n_in,
                              void* d_out, int out_size, void* d_ws, size_t ws_size,
                              hipStream_t stream) {
  const float* xA      = (const float*)d_in[0];
  const float* xB      = (const float*)d_in[1];
  const float* centerf = (const float*)d_in[2];
  const int*   labA    = (const int*)d_in[3];
  const int*   labB    = (const int*)d_in[4];
  float* out = (float*)d_out;

  const int D = 128;
  const int C = in_sizes[2] / D;   // 7
  const int N = in_sizes[3];       // 1,000,000

  float* c_pad = (float*)d_ws;     // 16*128 floats = 8 KB

  labelsim_prep_kernel<<<1, 128, 0, stream>>>(centerf, c_pad, out, C, D);

  const int tilesTot = 2 * ((N + 15) / 16);
  int blocks = (tilesTot + 7) / 8;           // 8 waves per 256-thread block
  if (blocks > 2048) blocks = 2048;
  if (blocks < 1) blocks = 1;
  labelsim_main_kernel<<<blocks, 256, 0, stream>>>(xA, xB, labA, labB,
                                                   c_pad, out, N);
}